// dynamic_routing_25280177504915
// MI455X (gfx1250) — compile-verified
//
#include <hip/hip_runtime.h>
#include <math.h>

#define BB   32
#define NN   8192
#define DIN  128
#define DOUT 128
#define EPSF 1e-8f

typedef __attribute__((ext_vector_type(2))) float v2f;
typedef __attribute__((ext_vector_type(8))) float v8f;

// ---------------------------------------------------------------------------
// Pass 0: rep = x_tile @ W via V_WMMA_F32_16X16X4_F32 (fp32 WMMA).
// Per block: 64 rows of one batch. 256 threads = 8 waves; each wave owns one
// 16-column block and 4 row-tiles (4 accumulators), so each B fragment of W
// is reused by 4 WMMAs -> 4x less W/L2 traffic, 4 independent WMMA chains.
// Produces: rep_norm[b,n], label[b,n], masked column sums of x + count
// (first iteration has b = 0/-inf, i.e. uniform softmax over non-padded rows,
//  so g1 = (1/count) * (Σ x_n) @ W).
// ---------------------------------------------------------------------------
__global__ void gemm_stats_kernel(const float* __restrict__ x,
                                  const float* __restrict__ W,
                                  float* __restrict__ rep_norm,
                                  float* __restrict__ label,
                                  float* __restrict__ s_accum,
                                  float* __restrict__ cnt)
{
    const int ROWS      = 64;                        // rows per block (4 M-tiles)
    const int tilesPerB = NN / ROWS;
    const int b    = blockIdx.x / tilesPerB;
    const int row0 = (blockIdx.x % tilesPerB) * ROWS;
    const int tid  = threadIdx.x;                    // 0..255

    __shared__ float Xs[64][132];    // 132 % 64 == 4 -> conflict-free column reads
    __shared__ float Rs[16][132];    // reused scratch for one 16-row rep tile
    __shared__ int   labS[64];

    // Cooperative load of the 64x128 x tile (float4, fully coalesced).
    const float* xg = x + ((size_t)b * NN + row0) * DIN;
    for (int idx = tid; idx < 64 * 32; idx += 256) {
        int r  = idx >> 5;
        int c4 = idx & 31;
        float4 v = ((const float4*)(xg + (size_t)r * DIN))[c4];
        Xs[r][c4 * 4 + 0] = v.x;
        Xs[r][c4 * 4 + 1] = v.y;
        Xs[r][c4 * 4 + 2] = v.z;
        Xs[r][c4 * 4 + 3] = v.w;
    }
    __syncthreads();

    const int wave = tid >> 5;
    const int lane = tid & 31;
    const int half = lane >> 4;      // 0: lanes 0-15, 1: lanes 16-31
    const int mn   = lane & 15;      // row (A) / col (B) index within tile
    const int n0   = wave * 16;      // this wave's output column block

    // 16x16x4 f32 WMMA fragments:
    //   A (16x4): lanes 0-15 hold row M=mn with K = k0,k0+1; lanes 16-31 K = k0+2,k0+3
    //   B (4x16): lanes 0-15 hold col N=mn with K = k0,k0+1; lanes 16-31 K = k0+2,k0+3
    v8f acc[4];
    #pragma unroll
    for (int t = 0; t < 4; ++t)
        acc[t] = (v8f){0.f, 0.f, 0.f, 0.f, 0.f, 0.f, 0.f, 0.f};

    #pragma unroll
    for (int k0 = 0; k0 < DIN; k0 += 4) {
        const int ka = k0 + half * 2;
        v2f bf;                                   // one B fragment, reused 4x
        bf.x = W[(size_t)ka       * DOUT + n0 + mn];
        bf.y = W[(size_t)(ka + 1) * DOUT + n0 + mn];
        #pragma unroll
        for (int t = 0; t < 4; ++t) {
            v2f a;
            a.x = Xs[t * 16 + mn][ka];
            a.y = Xs[t * 16 + mn][ka + 1];
            acc[t] = __builtin_amdgcn_wmma_f32_16x16x4_f32(
                /*neg_a=*/false, a, /*neg_b=*/false, bf,
                /*c_mod=*/(short)0, acc[t], /*reuse_a=*/false, /*reuse_b=*/false);
        }
    }

    // Per 16-row tile: scatter C fragment to LDS, then row stats.
    #pragma unroll
    for (int t = 0; t < 4; ++t) {
        __syncthreads();                          // Rs free (prev tile consumed)
        #pragma unroll
        for (int i = 0; i < 8; ++i)
            Rs[i + half * 8][n0 + mn] = acc[t][i];
        __syncthreads();
        if (tid < 16) {
            float s = 0.f, ss = 0.f;
            for (int c = 0; c < DOUT; ++c) { float v = Rs[tid][c]; s += v; ss += v * v; }
            int lab = (s == 0.0f);
            labS[t * 16 + tid] = lab;
            size_t gi = (size_t)b * NN + row0 + t * 16 + tid;
            rep_norm[gi] = fmaxf(sqrtf(ss), EPSF);
            label[gi]    = lab ? 1.0f : 0.0f;
        }
    }
    __syncthreads();

    // Masked column sums of x (uniform c in iteration 1) + non-padded count.
    if (tid < 128) {
        float a2 = 0.f;
        for (int r = 0; r < 64; ++r)
            if (!labS[r]) a2 += Xs[r][tid];
        atomicAdd(&s_accum[b * DIN + tid], a2);
    } else if (tid == 128) {
        int c = 0;
        for (int r = 0; r < 64; ++r) c += !labS[r];
        atomicAdd(&cnt[b], (float)c);
    }
}

// ---------------------------------------------------------------------------
// Finalize: g = (scale * s_accum) @ W ; g_norm = max(||g||, eps) ; u = W @ g.
// One block per batch, 128 threads.
// ---------------------------------------------------------------------------
__global__ void finalize_kernel(const float* __restrict__ s_accum,
                                const float* __restrict__ cnt,
                                const float* __restrict__ W,
                                float* __restrict__ g,
                                float* __restrict__ g_norm,
                                float* __restrict__ u,
                                int use_count)
{
    const int b = blockIdx.x;
    const int t = threadIdx.x;           // 0..127
    __shared__ float sh[128];
    __shared__ float gsh[128];
    __shared__ float red[128];

    float scale = 1.0f;
    if (use_count) scale = 1.0f / fmaxf(cnt[b], 1.0f);
    sh[t] = s_accum[b * DIN + t] * scale;
    __syncthreads();

    float go = 0.f;
    for (int i = 0; i < DIN; ++i) go += sh[i] * W[(size_t)i * DOUT + t];
    gsh[t] = go;
    g[b * DOUT + t] = go;
    red[t] = go * go;
    __syncthreads();
    for (int s = 64; s > 0; s >>= 1) {
        if (t < s) red[t] += red[t + s];
        __syncthreads();
    }
    if (t == 0) g_norm[b] = fmaxf(sqrtf(red[0]), EPSF);
    __syncthreads();

    float uu = 0.f;
    for (int o = 0; o < DOUT; ++o) uu += W[(size_t)t * DOUT + o] * gsh[o];
    u[b * DIN + t] = uu;
}

// ---------------------------------------------------------------------------
// Similarity pass: sim[b,n] = (x_n . u_b) / (rep_norm * g_norm), -inf if padded.
// One wave (32 lanes) per row; lane handles a float4 chunk of the 128-d dot.
// ---------------------------------------------------------------------------
__global__ void sim_kernel(const float* __restrict__ x,
                           const float* __restrict__ u,
                           const float* __restrict__ rep_norm,
                           const float* __restrict__ label,
                           const float* __restrict__ g_norm,
                           float* __restrict__ sim)
{
    const int ROWS = 64;
    const int blocksPerB = NN / ROWS;
    const int b    = blockIdx.x / blocksPerB;
    const int row0 = (blockIdx.x % blocksPerB) * ROWS;
    const int wave = threadIdx.x >> 5;
    const int lane = threadIdx.x & 31;

    float4 u4 = ((const float4*)(u + b * DIN))[lane];
    float  gn = g_norm[b];

    for (int r = wave; r < ROWS; r += 8) {
        size_t gi = (size_t)b * NN + row0 + r;
        float4 x4 = ((const float4*)(x + gi * DIN))[lane];
        float d = x4.x * u4.x + x4.y * u4.y + x4.z * u4.z + x4.w * u4.w;
        for (int off = 16; off > 0; off >>= 1) d += __shfl_xor(d, off, 32);
        if (lane == 0) {
            float s;
            if (label[gi] != 0.0f) s = -__builtin_inff();
            else                   s = d / (rep_norm[gi] * gn);
            sim[gi] = s;
        }
    }
}

// ---------------------------------------------------------------------------
// Softmax stats over N per batch: running max + sum of exp.
// ---------------------------------------------------------------------------
__global__ void softmax_stats_kernel(const float* __restrict__ sim,
                                     float* __restrict__ mx,
                                     float* __restrict__ den)
{
    const int b = blockIdx.x;
    const int t = threadIdx.x;           // 0..255
    __shared__ float red[256];
    __shared__ float msh;

    float m = -__builtin_inff();
    for (int n = t; n < NN; n += 256) m = fmaxf(m, sim[(size_t)b * NN + n]);
    red[t] = m;
    __syncthreads();
    for (int s = 128; s > 0; s >>= 1) {
        if (t < s) red[t] = fmaxf(red[t], red[t + s]);
        __syncthreads();
    }
    if (t == 0) { msh = red[0]; mx[b] = msh; }
    __syncthreads();

    const float M = msh;
    float sum = 0.f;
    for (int n = t; n < NN; n += 256) sum += expf(sim[(size_t)b * NN + n] - M);
    red[t] = sum;
    __syncthreads();
    for (int s = 128; s > 0; s >>= 1) {
        if (t < s) red[t] += red[t + s];
        __syncthreads();
    }
    if (t == 0) den[b] = red[0];
}

// ---------------------------------------------------------------------------
// Weighted column-sum of x: s_accum[b,:] += Σ_n softmax(sim)_n * x[b,n,:].
// ---------------------------------------------------------------------------
__global__ void wsum_kernel(const float* __restrict__ x,
                            const float* __restrict__ sim,
                            const float* __restrict__ mx,
                            const float* __restrict__ den,
                            float* __restrict__ s_accum)
{
    const int ROWS = 64;
    const int blocksPerB = NN / ROWS;
    const int b    = blockIdx.x / blocksPerB;
    const int row0 = (blockIdx.x % blocksPerB) * ROWS;
    const int t    = threadIdx.x;        // 0..127

    const float M    = mx[b];
    const float invd = 1.0f / den[b];

    float acc = 0.f;
    for (int r = 0; r < ROWS; ++r) {
        size_t gi = (size_t)b * NN + row0 + r;
        float c = expf(sim[gi] - M) * invd;   // sim = -inf -> c = 0
        acc += c * x[gi * DIN + t];
    }
    atomicAdd(&s_accum[b * DIN + t], acc);
}

// ---------------------------------------------------------------------------
extern "C" void kernel_launch(void* const* d_in, const int* in_sizes, int n_in,
                              void* d_out, int out_size, void* d_ws, size_t ws_size,
                              hipStream_t stream)
{
    (void)in_sizes; (void)n_in; (void)out_size; (void)ws_size;

    const float* x = (const float*)d_in[0];   // [32, 8192, 128] fp32
    const float* W = (const float*)d_in[1];   // [1, 128, 128]  fp32

    float* ws = (float*)d_ws;
    const size_t BN = (size_t)BB * NN;
    float* rep_norm = ws;                          // B*N
    float* label    = rep_norm + BN;               // B*N
    float* sim      = label + BN;                  // B*N
    float* s_accum  = sim + BN;                    // B*128
    float* cnt      = s_accum + (size_t)BB * DIN;  // B   (contiguous with s_accum)
    float* g        = cnt + BB;                    // B*128
    float* g_norm   = g + (size_t)BB * DOUT;       // B
    float* u        = g_norm + BB;                 // B*128
    float* mx       = u + (size_t)BB * DIN;        // B
    float* den      = mx + BB;                     // B

    // Iteration 1 (b = 0 / -inf  =>  uniform softmax over non-padded rows):
    hipMemsetAsync(s_accum, 0, ((size_t)BB * DIN + BB) * sizeof(float), stream);
    gemm_stats_kernel<<<BB * (NN / 64), 256, 0, stream>>>(x, W, rep_norm, label,
                                                          s_accum, cnt);
    finalize_kernel<<<BB, 128, 0, stream>>>(s_accum, cnt, W, g, g_norm, u, 1);

    // Iterations 2 and 3:
    for (int it = 0; it < 2; ++it) {
        sim_kernel<<<BB * (NN / 64), 256, 0, stream>>>(x, u, rep_norm, label,
                                                       g_norm, sim);
        softmax_stats_kernel<<<BB, 256, 0, stream>>>(sim, mx, den);
        hipMemsetAsync(s_accum, 0, (size_t)BB * DIN * sizeof(float), stream);
        wsum_kernel<<<BB * (NN / 64), 128, 0, stream>>>(x, sim, mx, den, s_accum);
        finalize_kernel<<<BB, 128, 0, stream>>>(s_accum, nullptr, W, g, g_norm, u, 0);
    }

    hipMemcpyAsync(d_out, g, (size_t)BB * DOUT * sizeof(float),
                   hipMemcpyDeviceToDevice, stream);
}